// ISNEModel_62113817035524
// MI455X (gfx1250) — compile-verified
//
#include <hip/hip_runtime.h>

#define EMBED_DIM 128
#define MAX_DEG 32
#define WAVES_PER_BLOCK 8
#define CHUNK 8   // neighbor rows per pipeline stage (double-buffered)

typedef float float4v  __attribute__((ext_vector_type(4)));
typedef unsigned int uint32x4 __attribute__((ext_vector_type(4)));
typedef int int32x4  __attribute__((ext_vector_type(4)));
typedef int int32x8  __attribute__((ext_vector_type(8)));

#if defined(__has_builtin)
#if __has_builtin(__builtin_amdgcn_tensor_load_to_lds) && \
    __has_builtin(__builtin_amdgcn_s_wait_tensorcnt)
#define USE_TDM_GATHER 1
#endif
#endif

// Raw LDS byte offset of a __shared__ object: generic -> addrspace(3) cast
// strips the aperture base; AS(3) pointers are 32-bit LDS offsets.
__device__ __forceinline__ unsigned lds_byte_offset(void* p) {
    return (unsigned)(unsigned long long)(__attribute__((address_space(3))) unsigned char*)p;
}

__device__ __forceinline__ void compiler_mem_fence() {
    asm volatile("" ::: "memory");
}

#if USE_TDM_GATHER
// ---- Tensor Data Mover gather-mode row DMA: theta rows -> LDS ------------
// One instruction gathers up to 8 rows (32-bit indices) of EMBED_DIM floats.
__device__ __forceinline__ void tdm_gather_rows(const float* __restrict__ theta,
                                                unsigned long long num_nodes,
                                                unsigned lds_addr,
                                                int my_nbr, int start, int todo) {
    const unsigned long long gaddr = (unsigned long long)theta;

    uint32x4 g0;
    g0.x = 0xC0000001u;                       // count=1, gather_mode=1, 32b idx
    g0.y = lds_addr;                          // LDS dest byte address
    g0.z = (unsigned)(gaddr & 0xFFFFFFFFu);   // global_addr[31:0]
    g0.w = (unsigned)(gaddr >> 32) | 0x80000000u;  // addr[56:32] | type=2

    int32x8 g1;
    g1[0] = (2 << 16);                                  // data_size = 4 bytes
    g1[1] = (EMBED_DIM & 0xFFFF) << 16;                 // tensor_dim0[15:0]
    g1[2] = (EMBED_DIM >> 16) |
            (int)((num_nodes & 0xFFFFull) << 16);       // dim0 hi | dim1 lo
    g1[3] = (int)(num_nodes >> 16) | (EMBED_DIM << 16); // dim1 hi | tile_dim0
    g1[4] = todo;                                       // tile_dim1 = #indices
    g1[5] = EMBED_DIM;                                  // dim0_stride (elements)
    g1[6] = 0;
    g1[7] = 0;

    // Row indices (groups 2/3), broadcast from lane-held neighbor ids.
    int32x4 g2, g3;
    g2[0] = __builtin_amdgcn_readlane(my_nbr, start + 0);
    g2[1] = __builtin_amdgcn_readlane(my_nbr, start + 1);
    g2[2] = __builtin_amdgcn_readlane(my_nbr, start + 2);
    g2[3] = __builtin_amdgcn_readlane(my_nbr, start + 3);
    g3[0] = __builtin_amdgcn_readlane(my_nbr, start + 4);
    g3[1] = __builtin_amdgcn_readlane(my_nbr, start + 5);
    g3[2] = __builtin_amdgcn_readlane(my_nbr, start + 6);
    g3[3] = __builtin_amdgcn_readlane(my_nbr, start + 7);

#if defined(__clang_major__) && __clang_major__ >= 23
    int32x8 z8 = {0, 0, 0, 0, 0, 0, 0, 0};
    __builtin_amdgcn_tensor_load_to_lds(g0, g1, g2, g3, z8, 0);
#else
    __builtin_amdgcn_tensor_load_to_lds(g0, g1, g2, g3, 0);
#endif
}

template <int N>
__device__ __forceinline__ void wait_chunk() {
    __builtin_amdgcn_s_wait_tensorcnt((short)N);
    compiler_mem_fence();
}
#else
// ---- Fallback: per-lane async DMA (ASYNCcnt) ------------------------------
__device__ __forceinline__ void async_copy_b128_to_lds(unsigned lds_off, const void* gaddr) {
    asm volatile("global_load_async_to_lds_b128 %0, %1, off"
                 :
                 : "v"(lds_off), "v"((unsigned long long)gaddr)
                 : "memory");
}

template <int N>
__device__ __forceinline__ void wait_chunk() {
    asm volatile("s_wait_asynccnt %0" :: "i"(N) : "memory");
}

__device__ __forceinline__ void issue_chunk_async(const float* __restrict__ theta,
                                                  int my_nbr, int start, int todo,
                                                  unsigned lds_base, int lane) {
    for (int s = 0; s < todo; ++s) {
        const int nk = __builtin_amdgcn_readlane(my_nbr, start + s);
        const float* row = theta + (long long)nk * EMBED_DIM;
        async_copy_b128_to_lds(lds_base + (unsigned)(s * EMBED_DIM * 4)
                                        + (unsigned)(lane * 16),
                               (const void*)(row + lane * 4));
    }
}
#endif

__global__ __launch_bounds__(WAVES_PER_BLOCK * 32)
void isne_gather_mean_kernel(const float* __restrict__ theta,
                             const int* __restrict__ node_ids,
                             const int* __restrict__ neighbors,
                             const int* __restrict__ degrees,
                             float* __restrict__ out,
                             int batch, int num_nodes) {
    // 8 waves x 2 buffers x 8 rows x 512 B = 64 KB / block
    __shared__ float stage[WAVES_PER_BLOCK][2][CHUNK * EMBED_DIM];

    const int lane = threadIdx.x & 31;
    const int wave = threadIdx.x >> 5;
    const int n = blockIdx.x * WAVES_PER_BLOCK + wave;
    if (n >= batch) return;

    // One wave owns one node -> degree is wave-uniform; force scalar control.
    const int deg = __builtin_amdgcn_readfirstlane(degrees[n]);
    // lane k holds neighbors[n][k]
    const int my_nbr = neighbors[(long long)n * MAX_DEG + lane];

    float4v acc = {0.f, 0.f, 0.f, 0.f};

    if (deg > 0) {
        const unsigned bases[2] = {
            static_cast<unsigned>(
                __builtin_amdgcn_readfirstlane(lds_byte_offset(&stage[wave][0][0]))),
            static_cast<unsigned>(
                __builtin_amdgcn_readfirstlane(lds_byte_offset(&stage[wave][1][0])))
        };
        int todo = (deg < CHUNK) ? deg : CHUNK;
#if USE_TDM_GATHER
        tdm_gather_rows(theta, (unsigned long long)num_nodes, bases[0], my_nbr, 0, todo);
#else
        issue_chunk_async(theta, my_nbr, 0, todo, bases[0], lane);
#endif

        int c = 0, buf = 0;
        for (;;) {
            const int next = c + CHUNK;
            int todo_next = deg - next;
            if (todo_next > CHUNK) todo_next = CHUNK;

            // Prefetch the next chunk into the other buffer before waiting.
            if (todo_next > 0) {
                compiler_mem_fence();
#if USE_TDM_GATHER
                tdm_gather_rows(theta, (unsigned long long)num_nodes,
                                bases[buf ^ 1], my_nbr, next, todo_next);
#else
                issue_chunk_async(theta, my_nbr, next, todo_next, bases[buf ^ 1], lane);
#endif
            }

            // Chunk DMAs complete in order: counter <= (next chunk's ops)
            // guarantees the current chunk has fully landed in LDS.
#if USE_TDM_GATHER
            if (todo_next > 0) wait_chunk<1>();
            else               wait_chunk<0>();
#else
            if (todo_next == CHUNK) wait_chunk<CHUNK>();
            else                    wait_chunk<0>();
#endif

            // Each lane reads its 4 dims per row: conflict-free ds_load_b128.
            for (int s = 0; s < todo; ++s) {
                float4v v = *(const float4v*)&stage[wave][buf][s * EMBED_DIM + lane * 4];
                acc += v;
            }
            // acc consumption drains DScnt before these LDS slots are re-targeted.

            if (todo_next <= 0) break;
            c = next; todo = todo_next; buf ^= 1;
        }
        acc *= (1.0f / (float)deg);
    } else {
        const int nid = __builtin_amdgcn_readfirstlane(node_ids[n]);
        acc = *(const float4v*)(theta + (long long)nid * EMBED_DIM + lane * 4);
    }

    *(float4v*)(out + (long long)n * EMBED_DIM + lane * 4) = acc;
}

extern "C" void kernel_launch(void* const* d_in, const int* in_sizes, int n_in,
                              void* d_out, int out_size, void* d_ws, size_t ws_size,
                              hipStream_t stream) {
    const float* theta     = (const float*)d_in[0];
    const int*   node_ids  = (const int*)d_in[1];
    const int*   neighbors = (const int*)d_in[2];
    const int*   degrees   = (const int*)d_in[3];
    float*       out       = (float*)d_out;

    const int batch     = in_sizes[1];               // node_ids element count
    const int num_nodes = in_sizes[0] / EMBED_DIM;   // theta rows
    const int blocks = (batch + WAVES_PER_BLOCK - 1) / WAVES_PER_BLOCK;
    isne_gather_mean_kernel<<<blocks, WAVES_PER_BLOCK * 32, 0, stream>>>(
        theta, node_ids, neighbors, degrees, out, batch, num_nodes);
}